// PositionalEncoding_8959301780112
// MI455X (gfx1250) — compile-verified
//
#include <hip/hip_runtime.h>
#include <hip/hip_bf16.h>
#include <stdint.h>

// Shapes from the reference
#define S_DIM 600
#define B_DIM 64
#define E_DIM 512
#define R_ROWS 1199          // 2*M - 1
#define CS_ROWS 1200         // prefix entries 0..1199
#define NGROUP 75            // 75 * 16 = 1200 >= 1199 rows

typedef __attribute__((ext_vector_type(2))) float v2f;
typedef __attribute__((ext_vector_type(4))) float v4f;
typedef __attribute__((ext_vector_type(8))) float v8f;

// ---------------------------------------------------------------------------
// Phase A: 16-row group column-sums of rel_table via V_WMMA_F32_16X16X4_F32.
// A = all-ones (16x4), B = 4x16 tile of rel_table; 4 accumulating WMMAs give
// D[0][n] = sum of 16 consecutive rows at column n. One wave per 16-col tile.
// Only one element (row 1199, kh==1, chunk 3, group 74) can be OOB, so only
// that single load is predicated; the other 7 are unconditional and issue
// back-to-back ahead of the WMMA chain.
// ---------------------------------------------------------------------------
__global__ __launch_bounds__(32) void pe_group_sums(const float* __restrict__ rel,
                                                    float* __restrict__ gsum) {
  const int lane = threadIdx.x;          // 0..31, full wave -> EXEC all ones
  const int n    = lane & 15;            // column within tile
  const int kh   = lane >> 4;            // K half: 0 -> K=0,1 ; 1 -> K=2,3
  const int g    = blockIdx.y;           // group 0..74
  const int col  = blockIdx.x * 16 + n;  // 0..511

  // base points at row (g*16 + kh*2), this lane's column
  const float* base = rel + (size_t)(g * 16 + kh * 2) * E_DIM + col;

  // Chunk c covers rows base + (4c)*E_DIM and base + (4c+1)*E_DIM.
  // Max unconditional row index: g*16 + 13 + kh*2 <= 74*16+15 = 1199 only for
  // the b3[1] slot; all others are <= 1198 < R_ROWS for every g.
  v2f b0, b1, b2, b3;
  b0[0] = base[0 * E_DIM];  b0[1] = base[1 * E_DIM];
  b1[0] = base[4 * E_DIM];  b1[1] = base[5 * E_DIM];
  b2[0] = base[8 * E_DIM];  b2[1] = base[9 * E_DIM];
  b3[0] = base[12 * E_DIM];
  float last = 0.0f;
  if (g * 16 + 13 + kh * 2 < R_ROWS) last = base[13 * E_DIM];
  b3[1] = last;

  v2f a;  a[0] = 1.0f; a[1] = 1.0f;      // ones in every lane => A is all-ones 16x4
  v8f acc = {};
  acc = __builtin_amdgcn_wmma_f32_16x16x4_f32(false, a, false, b0, (short)0, acc, false, false);
  acc = __builtin_amdgcn_wmma_f32_16x16x4_f32(false, a, false, b1, (short)0, acc, false, false);
  acc = __builtin_amdgcn_wmma_f32_16x16x4_f32(false, a, false, b2, (short)0, acc, false, false);
  acc = __builtin_amdgcn_wmma_f32_16x16x4_f32(false, a, false, b3, (short)0, acc, false, false);

  // D row 0 lives in VGPR0 of lanes 0..15
  if (kh == 0) gsum[(size_t)g * E_DIM + col] = acc[0];
}

// ---------------------------------------------------------------------------
// Phase B: exclusive prefix over the 75 group sums (tiny, 1 block).
// ---------------------------------------------------------------------------
__global__ __launch_bounds__(E_DIM) void pe_group_prefix(const float* __restrict__ gsum,
                                                         float* __restrict__ gpre) {
  const int e = threadIdx.x;
  float run = 0.0f;
  for (int g = 0; g < NGROUP; ++g) {
    gpre[g * E_DIM + e] = run;
    run += gsum[g * E_DIM + e];
  }
}

// ---------------------------------------------------------------------------
// Phase C: exact exclusive prefix cs[j][e] = sum of rel rows [0, j), j=0..1199.
// ---------------------------------------------------------------------------
__global__ __launch_bounds__(E_DIM) void pe_build_cs(const float* __restrict__ rel,
                                                     const float* __restrict__ gpre,
                                                     float* __restrict__ cs) {
  const int e = threadIdx.x;
  const int g = blockIdx.x;
  float run = gpre[g * E_DIM + e];
  #pragma unroll
  for (int r = 0; r < 16; ++r) {
    const int j = g * 16 + r;
    cs[(size_t)j * E_DIM + e] = run;
    if (j < R_ROWS) run += rel[(size_t)j * E_DIM + e];
  }
}

// ---------------------------------------------------------------------------
// Main streaming kernel: one block per s. Stage cs[s], cs[s+600] and the three
// temp_table rows (10 KB) into LDS with global_load_async_to_lds_b128, then
// stream x -> out in 16B vectors (4 b-lanes x 128 e4-lanes per 512-thread
// block). x / out are single-touch, so use non-temporal load/store to keep
// them out of L2; the reused broadcast data lives in LDS.
// ---------------------------------------------------------------------------
__global__ __launch_bounds__(512) void pe_main(const float* __restrict__ x,
                                               const int*   __restrict__ cfi,
                                               const float* __restrict__ cs,
                                               const float* __restrict__ temp,
                                               float* __restrict__ out) {
  __shared__ __align__(16) float sh[2560];  // [0,512) cs[s] | [512,1024) cs[s+600] | [1024,2560) temp rows
  __shared__ int shc[B_DIM];

  const int tid = threadIdx.x;
  const int s   = blockIdx.x;

  if (tid < B_DIM) shc[tid] = cfi[tid];

  // 640 16B chunks to stage: [0,128) cs row s, [128,256) cs row s+600, [256,640) temp (3x512 f32)
  const unsigned lds0 = (unsigned)(uintptr_t)(&sh[0]);   // low 32 bits = LDS byte offset
  {
    const int i = tid;                                    // chunks 0..511
    const float* g =
        (i < 128) ? (cs + (size_t)s * E_DIM + i * 4)
      : (i < 256) ? (cs + (size_t)(s + S_DIM) * E_DIM + (i - 128) * 4)
                  : (temp + (i - 256) * 4);
    const unsigned la = lds0 + (unsigned)(i * 16);
    const unsigned long long ga = (unsigned long long)(uintptr_t)g;
    asm volatile("global_load_async_to_lds_b128 %0, %1, off"
                 :: "v"(la), "v"(ga) : "memory");
  }
  if (tid < 128) {                                        // chunks 512..639 (rest of temp)
    const int i = 512 + tid;
    const float* g = temp + (i - 256) * 4;
    const unsigned la = lds0 + (unsigned)(i * 16);
    const unsigned long long ga = (unsigned long long)(uintptr_t)g;
    asm volatile("global_load_async_to_lds_b128 %0, %1, off"
                 :: "v"(la), "v"(ga) : "memory");
  }
  asm volatile("s_wait_asynccnt 0" ::: "memory");
  __syncthreads();

  const int e4   = tid & 127;   // 16B-vector index within an E row
  const int bsub = tid >> 7;    // 0..3

  const v4f ca = ((const v4f*)(sh       ))[e4];
  const v4f cb = ((const v4f*)(sh +  512))[e4];
  const v4f rel4 = cb - ca;
  const v4f t0 = ((const v4f*)(sh + 1024))[e4];
  const v4f t1 = ((const v4f*)(sh + 1536))[e4];
  const v4f t2 = ((const v4f*)(sh + 2048))[e4];

  const size_t rowbase = (size_t)s * (B_DIM * E_DIM);

  #pragma unroll 4
  for (int bb = 0; bb < B_DIM; bb += 4) {
    const int b = bb + bsub;
    const int c = shc[b];
    const v4f tv = (s < c) ? t2 : ((s == c) ? t0 : t1);
    const size_t off = rowbase + (size_t)b * E_DIM + (size_t)e4 * 4;
    const v4f xv = __builtin_nontemporal_load((const v4f*)(x + off));
    __builtin_nontemporal_store(xv + rel4 + tv, (v4f*)(out + off));
  }
}

// ---------------------------------------------------------------------------
// Launch: A -> B -> C -> main, all on `stream`. Workspace: cs | gsum | gpre.
// ---------------------------------------------------------------------------
extern "C" void kernel_launch(void* const* d_in, const int* in_sizes, int n_in,
                              void* d_out, int out_size, void* d_ws, size_t ws_size,
                              hipStream_t stream) {
  const float* x    = (const float*)d_in[0];
  const int*   cfi  = (const int*)  d_in[1];
  const float* rel  = (const float*)d_in[2];
  const float* temp = (const float*)d_in[3];
  float*       out  = (float*)d_out;

  float* cs   = (float*)d_ws;                       // 1200 * 512 f32
  float* gsum = cs   + (size_t)CS_ROWS * E_DIM;     //   75 * 512 f32
  float* gpre = gsum + (size_t)NGROUP  * E_DIM;     //   75 * 512 f32

  pe_group_sums  <<<dim3(E_DIM / 16, NGROUP), 32, 0, stream>>>(rel, gsum);
  pe_group_prefix<<<1,      E_DIM, 0, stream>>>(gsum, gpre);
  pe_build_cs    <<<NGROUP, E_DIM, 0, stream>>>(rel, gpre, cs);
  pe_main        <<<S_DIM,  512,   0, stream>>>(x, cfi, cs, temp, out);
}